// WeightedSVD_31275951849976
// MI455X (gfx1250) — compile-verified
//
#include <hip/hip_runtime.h>
#include <math.h>

// ---------------------------------------------------------------------------
// WeightedSVD on MI455X (gfx1250):
//   per batch b: M = Zh^T diag(W) Zh (3x3 sym PSD), output = eigvec of the
//   smallest eigenvalue of M (== last left singular vector of SVD(M)).
//
// Memory-bound problem: ~201 MB streamed once @ 23.3 TB/s => ~8.6us floor.
// Strategy: 1 workgroup per batch, 256 threads, vectorized streaming loads,
// LDS tree reduction to 4 group-partials, final 4-way reduction done with a
// single V_WMMA_F32_16X16X4_F32 (A = 6x4 partial matrix, B = ones), then a
// closed-form 3x3 symmetric eigensolve in f64 on lane 0.
// ---------------------------------------------------------------------------

typedef __attribute__((ext_vector_type(2))) float v2f;
typedef __attribute__((ext_vector_type(8))) float v8f;

#define THREADS 256

__global__ __launch_bounds__(THREADS)
void wsvd_kernel(const float* __restrict__ Z,   // (B, N, 2)
                 const float* __restrict__ W,   // (B, N)
                 float* __restrict__ out,       // (B, 3)
                 int n) {
    const int b = blockIdx.x;
    const int t = threadIdx.x;
    const float* zb = Z + (size_t)b * (size_t)n * 2;
    const float* wb = W + (size_t)b * (size_t)n;

    // ---- Phase 1: streaming accumulation (2 points / thread / iteration) ----
    float sxx = 0.f, sxy = 0.f, syy = 0.f, sx = 0.f, sy = 0.f, sw = 0.f;
    for (int base = 2 * t; base + 1 < n; base += 2 * THREADS) {
        // prefetch ~2 iterations ahead (global_prefetch_b8 on gfx1250)
        if (base + 4 * THREADS + 1 < n) {
            __builtin_prefetch(zb + 2 * (base + 4 * THREADS), 0, 0);
            __builtin_prefetch(wb + (base + 4 * THREADS), 0, 0);
        }
        float4 z2 = *(const float4*)(zb + 2 * base);   // global_load_b128
        float2 w2 = *(const float2*)(wb + base);       // global_load_b64

        float wx0 = w2.x * z2.x, wy0 = w2.x * z2.y;
        sxx += wx0 * z2.x;  sxy += wx0 * z2.y;  syy += wy0 * z2.y;
        sx  += wx0;         sy  += wy0;         sw  += w2.x;

        float wx1 = w2.y * z2.z, wy1 = w2.y * z2.w;
        sxx += wx1 * z2.z;  sxy += wx1 * z2.w;  syy += wy1 * z2.w;
        sx  += wx1;         sy  += wy1;         sw  += w2.y;
    }

    // ---- Phase 2: LDS tree reduction into 4 groups of 64 threads ----
    __shared__ float lds[THREADS][6];
    lds[t][0] = sxx; lds[t][1] = sxy; lds[t][2] = syy;
    lds[t][3] = sx;  lds[t][4] = sy;  lds[t][5] = sw;
    __syncthreads();

    for (int off = 32; off >= 1; off >>= 1) {
        if ((t & 63) < off) {
            #pragma unroll
            for (int c = 0; c < 6; ++c) lds[t][c] += lds[t + off][c];
        }
        __syncthreads();
    }
    // group sums now live at lds[0][*], lds[64][*], lds[128][*], lds[192][*]

    // ---- Phase 3: final 4-way reduction via V_WMMA_F32_16X16X4_F32 ----
    // A(m,k) = partial sum of quantity m (0..5) over thread-group k (0..3),
    //          rows 6..15 zero. Documented f32 16x4 A layout:
    //            VGPR0: K=0 (lanes 0-15), K=2 (lanes 16-31)
    //            VGPR1: K=1 (lanes 0-15), K=3 (lanes 16-31)
    // B = ones(4x16)  -> layout-independent.
    // D(m,n) = sum_k A(m,k); lane 0 holds D(m,0) in VGPR m (m = 0..7).
    if (t < 32) {
        const int q  = t & 15;     // A row (quantity index)
        const int hi = t >> 4;     // lane half selects groups {0,1} vs {2,3}
        v2f a;
        a[0] = (q < 6) ? lds[hi ? 128 : 0][q]  : 0.f;  // K = 0 or 2
        a[1] = (q < 6) ? lds[hi ? 192 : 64][q] : 0.f;  // K = 1 or 3
        v2f ones;
        ones[0] = 1.0f; ones[1] = 1.0f;
        v8f cacc = {};
        v8f d = __builtin_amdgcn_wmma_f32_16x16x4_f32(
            /*neg_a=*/false, a, /*neg_b=*/false, ones,
            /*c_mod=*/(short)0, cacc, /*reuse_a=*/false, /*reuse_b=*/false);

        // ---- Phase 4: 3x3 symmetric eigensolve (smallest eigenvalue) ----
        if (t == 0) {
            const double m00 = d[0];  // Sxx
            const double m01 = d[1];  // Sxy
            const double m11 = d[2];  // Syy
            const double m02 = d[3];  // Sx
            const double m12 = d[4];  // Sy
            const double m22 = d[5];  // Sw

            // Trigonometric eigenvalue formula (symmetric 3x3)
            const double p1 = m01 * m01 + m02 * m02 + m12 * m12;
            const double qv = (m00 + m11 + m22) * (1.0 / 3.0);
            const double a00 = m00 - qv, a11 = m11 - qv, a22 = m22 - qv;
            const double p2 = a00 * a00 + a11 * a11 + a22 * a22 + 2.0 * p1;
            const double pv = sqrt(p2 * (1.0 / 6.0));
            double lam = qv;
            if (pv > 1e-300) {
                const double inv = 1.0 / pv;
                const double b00 = a00 * inv, b01 = m01 * inv, b02 = m02 * inv;
                const double b11 = a11 * inv, b12 = m12 * inv, b22 = a22 * inv;
                double r = 0.5 * (b00 * (b11 * b22 - b12 * b12)
                                - b01 * (b01 * b22 - b12 * b02)
                                + b02 * (b01 * b12 - b11 * b02));
                r = fmin(1.0, fmax(-1.0, r));
                const double phi = acos(r) * (1.0 / 3.0);
                // lambda_min = q + 2p*cos(phi + 2*pi/3)
                lam = qv + 2.0 * pv * cos(phi + 2.0943951023931954828);
            }

            // Eigenvector: most-independent cross product of rows of (M - lam*I)
            const double r0x = m00 - lam, r0y = m01,       r0z = m02;
            const double r1x = m01,       r1y = m11 - lam, r1z = m12;
            const double r2x = m02,       r2y = m12,       r2z = m22 - lam;

            double c0x = r0y * r1z - r0z * r1y;
            double c0y = r0z * r1x - r0x * r1z;
            double c0z = r0x * r1y - r0y * r1x;
            double c1x = r0y * r2z - r0z * r2y;
            double c1y = r0z * r2x - r0x * r2z;
            double c1z = r0x * r2y - r0y * r2x;
            double c2x = r1y * r2z - r1z * r2y;
            double c2y = r1z * r2x - r1x * r2z;
            double c2z = r1x * r2y - r1y * r2x;

            const double n0 = c0x * c0x + c0y * c0y + c0z * c0z;
            const double n1 = c1x * c1x + c1y * c1y + c1z * c1z;
            const double n2 = c2x * c2x + c2y * c2y + c2z * c2z;

            double vx = c0x, vy = c0y, vz = c0z, nn = n0;
            if (n1 > nn) { vx = c1x; vy = c1y; vz = c1z; nn = n1; }
            if (n2 > nn) { vx = c2x; vy = c2y; vz = c2z; nn = n2; }
            if (!(nn > 1e-300)) { vx = 0.0; vy = 0.0; vz = 1.0; nn = 1.0; }

            const double s = 1.0 / sqrt(nn);
            out[(size_t)b * 3 + 0] = (float)(vx * s);
            out[(size_t)b * 3 + 1] = (float)(vy * s);
            out[(size_t)b * 3 + 2] = (float)(vz * s);
        }
    }
}

extern "C" void kernel_launch(void* const* d_in, const int* in_sizes, int n_in,
                              void* d_out, int out_size, void* d_ws, size_t ws_size,
                              hipStream_t stream) {
    (void)n_in; (void)d_ws; (void)ws_size;
    const float* Z = (const float*)d_in[0];   // (B, N, 2) f32
    const float* W = (const float*)d_in[1];   // (B, N)    f32
    float* out = (float*)d_out;               // (B, 3)    f32

    const int B = out_size / 3;               // 2048
    const int N = in_sizes[1] / B;            // 8192

    wsvd_kernel<<<dim3(B), dim3(THREADS), 0, stream>>>(Z, W, out, N);
}